// VolumeRenderer_45217415693185
// MI455X (gfx1250) — compile-verified
//
#include <hip/hip_runtime.h>
#include <stdint.h>

#define NSAMP 128
#define WAVES_PER_BLOCK 8
#define THREADS (WAVES_PER_BLOCK * 32)

typedef float v4f __attribute__((ext_vector_type(4)));

// ---------------------------------------------------------------------------
// CDNA5 async global->LDS copy (16B per lane), tracked by ASYNCcnt.
// VDST = VGPR holding LDS byte offset, VADDR = VGPR pair w/ 64-bit address.
// Inputs are single-use -> non-temporal load hint (keep GL2 for the stream).
// ---------------------------------------------------------------------------
__device__ __forceinline__ void async_copy_b128(unsigned lds_byte_off,
                                                const float* gsrc) {
  asm volatile("global_load_async_to_lds_b128 %0, %1, off th:TH_LOAD_NT"
               :: "v"(lds_byte_off), "v"(gsrc)
               : "memory");
}

__device__ __forceinline__ void prefetch_ray(unsigned buf_byte_off,
                                             const float* __restrict__ rgb,
                                             const float* __restrict__ sigma,
                                             int ray, unsigned lane) {
  // sigma: 128 floats -> LDS [buf + 0, buf + 512)
  const float* sg = sigma + (size_t)ray * NSAMP + lane * 4u;
  async_copy_b128(buf_byte_off + lane * 16u, sg);
  // rgb: 384 floats -> LDS [buf + 512, buf + 2048). lane owns 12 floats.
  const float* rg = rgb + (size_t)ray * (NSAMP * 3) + lane * 12u;
  const unsigned rb = buf_byte_off + 512u + lane * 48u;
  async_copy_b128(rb,        rg);
  async_copy_b128(rb + 16u,  rg + 4);
  async_copy_b128(rb + 32u,  rg + 8);
}

__global__ __launch_bounds__(THREADS) void volrender_kernel(
    const float* __restrict__ rays_d,
    const float* __restrict__ rgb,
    const float* __restrict__ sigma,
    float* __restrict__ out_rgb,    // [N,3]
    float* __restrict__ out_depth,  // [N]
    float* __restrict__ out_w,      // [N,128]
    float* __restrict__ out_z,      // [N,128]
    int n_rays, int total_waves) {
  // Per wave: 2 buffers x (128 sigma + 384 rgb) floats = 1024 floats = 4KB.
  __shared__ float smem[WAVES_PER_BLOCK * 1024];

  const unsigned lane  = threadIdx.x & 31u;
  const unsigned wv    = threadIdx.x >> 5;
  const int      gwave = (int)(blockIdx.x * WAVES_PER_BLOCK + wv);
  const unsigned wbase = wv * 1024u;   // float index of this wave's LDS region
  const unsigned wbyte = wbase * 4u;   // byte offset (smem starts at LDS offset 0)

  const float inv127 = 1.0f / 127.0f;

  if (gwave < n_rays) {
    prefetch_ray(wbyte, rgb, sigma, gwave, lane);
  }

  unsigned parity = 0;
  for (int ray = gwave; ray < n_rays; ray += total_waves) {
    const int nxt = ray + total_waves;
    if (nxt < n_rays) {
      prefetch_ray(wbyte + (parity ^ 1u) * 2048u, rgb, sigma, nxt, lane);
      // 4 newest in flight belong to the next ray; everything older is done.
      asm volatile("s_wait_asynccnt 0x4" ::: "memory");
    } else {
      asm volatile("s_wait_asynccnt 0x0" ::: "memory");
    }

    const unsigned cbase = wbase + parity * 512u;  // float index of cur buffer

    // |rays_d| -- all lanes load the same 3 floats (broadcast).
    const float dx = rays_d[3 * (size_t)ray + 0];
    const float dy = rays_d[3 * (size_t)ray + 1];
    const float dz = rays_d[3 * (size_t)ray + 2];
    const float nrm = sqrtf(dx * dx + dy * dy + dz * dz);

    // This lane's 4 samples: s = 4*lane + k
    const v4f s4 = *(const v4f*)&smem[cbase + lane * 4u];
    const int s0 = (int)lane * 4;
    const float sig[4] = {s4.x, s4.y, s4.z, s4.w};

    float a[4], f[4], zv[4];
#pragma unroll
    for (int k = 0; k < 4; ++k) {
      const int s = s0 + k;
      const float dist = (s == NSAMP - 1) ? 1e10f : inv127;
      const float e = expf(-sig[k] * (dist * nrm));
      a[k] = 1.0f - e;            // alpha
      f[k] = e + 1e-10f;          // 1 - alpha + eps
      zv[k] = (s == NSAMP - 1) ? 1.0f : (float)s * inv127;
    }

    // In-lane exclusive prefix products of f.
    const float t0 = 1.0f;
    const float t1 = f[0];
    const float t2 = f[0] * f[1];
    const float t3 = t2 * f[2];
    const float plocal = t3 * f[3];

    // Wave32 inclusive product scan over lane-local products.
    float incl = plocal;
#pragma unroll
    for (int off = 1; off < 32; off <<= 1) {
      const float v = __shfl_up(incl, off, 32);
      if ((int)lane >= off) incl *= v;
    }
    float excl = __shfl_up(incl, 1, 32);
    if (lane == 0) excl = 1.0f;

    const float w0 = a[0] * (excl * t0);
    const float w1 = a[1] * (excl * t1);
    const float w2 = a[2] * (excl * t2);
    const float w3 = a[3] * (excl * t3);

    // rgb for this lane's 4 samples: 12 contiguous floats from LDS.
    const v4f* rp = (const v4f*)&smem[cbase + 128u + lane * 12u];
    const v4f c0 = rp[0];
    const v4f c1 = rp[1];
    const v4f c2 = rp[2];
    // sample k -> channels: k0:(c0.x,c0.y,c0.z) k1:(c0.w,c1.x,c1.y)
    //                       k2:(c1.z,c1.w,c2.x) k3:(c2.y,c2.z,c2.w)
    float rr = w0 * c0.x + w1 * c0.w + w2 * c1.z + w3 * c2.y;
    float gg = w0 * c0.y + w1 * c1.x + w2 * c1.w + w3 * c2.z;
    float bb = w0 * c0.z + w1 * c1.y + w2 * c2.x + w3 * c2.w;
    float dd = w0 * zv[0] + w1 * zv[1] + w2 * zv[2] + w3 * zv[3];

    // Wave32 butterfly reduction (sums end up in every lane).
#pragma unroll
    for (int m = 16; m >= 1; m >>= 1) {
      rr += __shfl_xor(rr, m, 32);
      gg += __shfl_xor(gg, m, 32);
      bb += __shfl_xor(bb, m, 32);
      dd += __shfl_xor(dd, m, 32);
    }

    // weights / z_vals: 512MB of single-use output -> non-temporal stores.
    v4f wq; wq.x = w0; wq.y = w1; wq.z = w2; wq.w = w3;
    __builtin_nontemporal_store(wq, (v4f*)(out_w + (size_t)ray * NSAMP + lane * 4u));
    v4f zq; zq.x = zv[0]; zq.y = zv[1]; zq.z = zv[2]; zq.w = zv[3];
    __builtin_nontemporal_store(zq, (v4f*)(out_z + (size_t)ray * NSAMP + lane * 4u));

    // Per-ray scalars: spread across lanes 0..3 (one store each).
    if (lane < 3) {
      const float c = (lane == 0) ? rr : ((lane == 1) ? gg : bb);
      out_rgb[3 * (size_t)ray + lane] = c;
    } else if (lane == 3) {
      out_depth[ray] = dd;
    }

    parity ^= 1u;
  }
}

extern "C" void kernel_launch(void* const* d_in, const int* in_sizes, int n_in,
                              void* d_out, int out_size, void* d_ws, size_t ws_size,
                              hipStream_t stream) {
  // setup_inputs order: rays_o [N,3] (unused), rays_d [N,3], rgb [N,128,3],
  // sigma [N,128,1]. Outputs concatenated: rgb_map N*3 | depth N | weights
  // N*128 | z_vals N*128.
  const float* rays_d = (const float*)d_in[1];
  const float* rgb    = (const float*)d_in[2];
  const float* sigma  = (const float*)d_in[3];
  const int N = in_sizes[0] / 3;

  float* out       = (float*)d_out;
  float* out_rgb   = out;
  float* out_depth = out + (size_t)N * 3;
  float* out_w     = out + (size_t)N * 4;
  float* out_z     = out_w + (size_t)N * NSAMP;

  int blocks = 4096;  // 32768 waves; 8 rays per wave -> async double-buffer pays off
  const int max_blocks = (N + WAVES_PER_BLOCK - 1) / WAVES_PER_BLOCK;
  if (blocks > max_blocks) blocks = max_blocks;
  const int total_waves = blocks * WAVES_PER_BLOCK;

  volrender_kernel<<<blocks, THREADS, 0, stream>>>(
      rays_d, rgb, sigma, out_rgb, out_depth, out_w, out_z, N, total_waves);
}